// dpc_knn_1795296330197
// MI455X (gfx1250) — compile-verified
//
#include <hip/hip_runtime.h>
#include <hip/hip_bf16.h>
#include <math.h>

// ---------------------------------------------------------------------------
// DPC-KNN token clustering for MI455X (gfx1250, wave32).
// Batched 196x768 @ 768x196 Gram matrix via V_WMMA_F32_16X16X4_F32 (full fp32;
// the problem is bandwidth bound so low-precision WMMA buys nothing).
// GEMM is LDS-tiled: 64x64 output block per 8-wave workgroup, K staged in
// 96-float chunks with padded stride (bank-conflict-free ds_load_b64 fragment
// reads, 16B-aligned ds_store_b128 staging). Cuts L2 traffic ~2.7x vs the
// wave-per-tile direct-gather version.
// ---------------------------------------------------------------------------

static const int N_TOK = 196;   // 14*14 patch tokens
static const int CF    = 768;   // 16*16*3 features per token
static const int KEEP  = 49;    // len_keep (hard-coded in reference setup)
static const int BKC   = 96;    // K chunk staged in LDS
static const int LSTR  = 100;   // padded LDS row stride (floats)

typedef __attribute__((ext_vector_type(2))) float v2f;
typedef __attribute__((ext_vector_type(8))) float v8f;

// ---- 1. rearrange b c (14 h) (14 w) -> b (14*14) (h w c), input-coalesced --
__global__ __launch_bounds__(256)
void tokenize_kernel(const float* __restrict__ x, float* __restrict__ tok,
                     long total) {
  long id = (long)blockIdx.x * blockDim.x + threadIdx.x;
  if (id >= total) return;
  int wx = (int)(id % 224);
  int hy = (int)((id / 224) % 224);
  int c  = (int)((id / (224 * 224)) % 3);
  long b = id / (3L * 224 * 224);
  int n = (hy >> 4) * 14 + (wx >> 4);                 // token index
  int f = (((hy & 15) << 4) + (wx & 15)) * 3 + c;     // feature index
  tok[(b * N_TOK + n) * (long)CF + f] = x[id];
}

// ---- 2. squared norms per token -------------------------------------------
__global__ __launch_bounds__(256)
void sqnorm_kernel(const float* __restrict__ tok, float* __restrict__ sq) {
  __shared__ float red[256];
  long base = (long)blockIdx.x * CF;
  int t = threadIdx.x;
  float a0 = tok[base + t];
  float a1 = tok[base + t + 256];
  float a2 = tok[base + t + 512];
  red[t] = a0 * a0 + a1 * a1 + a2 * a2;
  __syncthreads();
  for (int s = 128; s > 0; s >>= 1) {
    if (t < s) red[t] += red[t + s];
    __syncthreads();
  }
  if (t == 0) sq[blockIdx.x] = red[0];
}

// ---- 3. batched Gram via fp32 WMMA with LDS tiling ------------------------
// Block = 8 waves = 64x64 output super-tile (4x4 WMMA tiles, 2 tiles/wave).
// dist = sqrt(max(|ti|^2+|tj|^2-2*ti.tj, 0)) / sqrt(CF)
__global__ __launch_bounds__(256)
void gemm_dist_kernel(const float* __restrict__ tok,
                      const float* __restrict__ sq,
                      float* __restrict__ dist) {
  __shared__ float lA[64 * LSTR];
  __shared__ float lB[64 * LSTR];
  int bid = blockIdx.x;
  int b   = bid >> 4;                 // 16 blocks per batch (4x4 of 64x64)
  int bm  = (bid >> 2) & 3;
  int bn  = bid & 3;
  const float* T = tok + (long)b * N_TOK * CF;

  int tid  = threadIdx.x;
  int lane = tid & 31;
  int wave = tid >> 5;
  int half = lane >> 4;               // lanes 16-31 hold K+2..K+3 of fragment
  int l16  = lane & 15;
  int tr   = wave & 3;                // tile row within block
  int tc0  = (wave >> 2) << 1;        // tile col pair {0,1} or {2,3}

  v8f acc0 = {0.f, 0.f, 0.f, 0.f, 0.f, 0.f, 0.f, 0.f};
  v8f acc1 = {0.f, 0.f, 0.f, 0.f, 0.f, 0.f, 0.f, 0.f};

  for (int k0 = 0; k0 < CF; k0 += BKC) {
    // stage 64 A-rows and 64 B-rows, cols [k0, k0+96), coalesced float4 loads
    for (int i = tid; i < 64 * (BKC / 4); i += 256) {
      int row = i / (BKC / 4);
      int c4  = i % (BKC / 4);
      int ga = bm * 64 + row; ga = ga < N_TOK ? ga : N_TOK - 1;
      int gb = bn * 64 + row; gb = gb < N_TOK ? gb : N_TOK - 1;
      float4 va = *(const float4*)(T + (long)ga * CF + k0 + c4 * 4);
      float4 vb = *(const float4*)(T + (long)gb * CF + k0 + c4 * 4);
      *(float4*)&lA[row * LSTR + c4 * 4] = va;
      *(float4*)&lB[row * LSTR + c4 * 4] = vb;
    }
    __syncthreads();
#pragma unroll 4
    for (int k = 0; k < BKC; k += 4) {
      v2f a  = *(const v2f*)&lA[(tr * 16 + l16) * LSTR + k + 2 * half];
      v2f b0 = *(const v2f*)&lB[(tc0 * 16 + l16) * LSTR + k + 2 * half];
      v2f b1 = *(const v2f*)&lB[((tc0 + 1) * 16 + l16) * LSTR + k + 2 * half];
      acc0 = __builtin_amdgcn_wmma_f32_16x16x4_f32(
          false, a, false, b0, (short)0, acc0, false, false);
      acc1 = __builtin_amdgcn_wmma_f32_16x16x4_f32(
          false, a, false, b1, (short)0, acc1, false, false);
    }
    __syncthreads();
  }

  const float inv = 0.03608439182435161f;   // 1/sqrt(768)
  const float* sqv = sq + (long)b * N_TOK;
  float* D = dist + (long)b * N_TOK * N_TOK;
  int n0 = bn * 64 + tc0 * 16 + l16;
  int n1 = n0 + 16;
  float s0 = (n0 < N_TOK) ? sqv[n0] : 0.f;
  float s1 = (n1 < N_TOK) ? sqv[n1] : 0.f;
#pragma unroll
  for (int r = 0; r < 8; ++r) {
    int M = bm * 64 + tr * 16 + r + 8 * half;   // C/D layout: VGPR r + half
    if (M < N_TOK) {
      float sm = sqv[M];
      if (n0 < N_TOK) {
        float d2 = sm + s0 - 2.0f * acc0[r];
        D[(long)M * N_TOK + n0] = sqrtf(fmaxf(d2, 0.f)) * inv;
      }
      if (n1 < N_TOK) {
        float d2 = sm + s1 - 2.0f * acc1[r];
        D[(long)M * N_TOK + n1] = sqrtf(fmaxf(d2, 0.f)) * inv;
      }
    }
  }
}

// ---- 4. kNN(5) local density per token ------------------------------------
__global__ __launch_bounds__(256)
void density_kernel(const float* __restrict__ dist, float* __restrict__ dens,
                    int nbn) {
  int id = blockIdx.x * blockDim.x + threadIdx.x;
  if (id >= nbn) return;
  const float* row = dist + (long)id * N_TOK;
  float t0 = 1e30f, t1 = 1e30f, t2 = 1e30f, t3 = 1e30f, t4 = 1e30f;
  for (int m = 0; m < N_TOK; ++m) {
    float v = row[m];
    if (v < t4) {
      t4 = v;
      if (t4 < t3) { float tt = t3; t3 = t4; t4 = tt;
        if (t3 < t2) { tt = t2; t2 = t3; t3 = tt;
          if (t2 < t1) { tt = t1; t1 = t2; t2 = tt;
            if (t1 < t0) { tt = t0; t0 = t1; t1 = tt; } } } }
    }
  }
  float ms = (t0*t0 + t1*t1 + t2*t2 + t3*t3 + t4*t4) * 0.2f;
  dens[id] = expf(-ms);
}

// ---- 5. per-batch max distance --------------------------------------------
__global__ __launch_bounds__(256)
void distmax_kernel(const float* __restrict__ dist, float* __restrict__ dmax) {
  __shared__ float red[256];
  const float* D = dist + (long)blockIdx.x * N_TOK * N_TOK;
  int t = threadIdx.x;
  float m = 0.f;
  for (int i = t; i < N_TOK * N_TOK; i += 256) m = fmaxf(m, D[i]);
  red[t] = m;
  __syncthreads();
  for (int s = 128; s > 0; s >>= 1) {
    if (t < s) red[t] = fmaxf(red[t], red[t + s]);
    __syncthreads();
  }
  if (t == 0) dmax[blockIdx.x] = red[0];
}

// ---- 6. dist to nearest higher-density token; score = dmin * density ------
__global__ __launch_bounds__(256)
void score_kernel(const float* __restrict__ dist, const float* __restrict__ dens,
                  const float* __restrict__ dmax, float* __restrict__ score,
                  int nbn) {
  int id = blockIdx.x * blockDim.x + threadIdx.x;
  if (id >= nbn) return;
  int b = id / N_TOK, n = id % N_TOK;
  const float* row = dist + (long)id * N_TOK;
  const float* dv = dens + (long)b * N_TOK;
  float dn = dv[n];
  float best = dmax[b];
  for (int m = 0; m < N_TOK; ++m) {
    float dm = dv[m];
    // index tie-break stands in for the reference's 1e-6 uniform noise
    bool higher = (dm > dn) || (dm == dn && m < n);
    if (higher) best = fminf(best, row[m]);
  }
  score[id] = best * dn;
}

// ---- 7. top-49 scores per batch (one wave32 per batch) --------------------
__global__ __launch_bounds__(32)
void topk_kernel(const float* __restrict__ score, int* __restrict__ index_down) {
  int b = blockIdx.x;
  int lane = threadIdx.x;
  float vals[7];
  int   idxs[7];
#pragma unroll
  for (int j = 0; j < 7; ++j) {
    int n = lane + 32 * j;
    idxs[j] = n;
    vals[j] = (n < N_TOK) ? score[(long)b * N_TOK + n] : -1e30f;
  }
  for (int k = 0; k < KEEP; ++k) {
    float bv = -1e30f; int bi = 1 << 30;
#pragma unroll
    for (int j = 0; j < 7; ++j) {
      if (vals[j] > bv || (vals[j] == bv && idxs[j] < bi)) { bv = vals[j]; bi = idxs[j]; }
    }
    float rv = bv; int ri = bi;
    for (int off = 16; off > 0; off >>= 1) {
      float ov = __shfl_xor(rv, off);
      int   oi = __shfl_xor(ri, off);
      if (ov > rv || (ov == rv && oi < ri)) { rv = ov; ri = oi; }
    }
    if (lane == 0) index_down[(long)b * KEEP + k] = ri;
    if ((ri & 31) == lane) vals[ri >> 5] = -1e30f;   // winner lane retires it
  }
}

// ---- 8. assign each token to nearest center -------------------------------
__global__ __launch_bounds__(256)
void assign_kernel(const float* __restrict__ dist, const int* __restrict__ index_down,
                   int* __restrict__ idx_cluster, int nbn) {
  int id = blockIdx.x * blockDim.x + threadIdx.x;
  if (id >= nbn) return;
  int b = id / N_TOK, n = id % N_TOK;
  const float* D = dist + (long)b * N_TOK * N_TOK;
  const int* cent = index_down + (long)b * KEEP;
  float best = 1e30f; int bi = 0;
  for (int k = 0; k < KEEP; ++k) {
    float d = D[(long)cent[k] * N_TOK + n];
    if (d < best) { best = d; bi = k; }      // strict < == argmin first-hit
  }
  idx_cluster[id] = bi;
}

// ---- 9. centers claim their own cluster id --------------------------------
__global__ __launch_bounds__(256)
void center_fix_kernel(const int* __restrict__ index_down,
                       int* __restrict__ idx_cluster, int nbk) {
  int id = blockIdx.x * blockDim.x + threadIdx.x;
  if (id >= nbk) return;
  int b = id / KEEP, k = id % KEEP;
  idx_cluster[(long)b * N_TOK + index_down[id]] = k;
}

// ---- 10. representative per cluster (stable-sort cumsum pick == last member)
__global__ __launch_bounds__(256)
void rep_kernel(const int* __restrict__ idx_cluster, int* __restrict__ out,
                int nbk) {
  int id = blockIdx.x * blockDim.x + threadIdx.x;
  if (id >= nbk) return;
  int b = id / KEEP, k = id % KEEP;
  const int* cl = idx_cluster + (long)b * N_TOK;
  int last = 0;
  for (int n = 0; n < N_TOK; ++n)
    if (cl[n] == k) last = n;
  out[id] = last;
}

// ---------------------------------------------------------------------------
extern "C" void kernel_launch(void* const* d_in, const int* in_sizes, int n_in,
                              void* d_out, int out_size, void* d_ws, size_t ws_size,
                              hipStream_t stream) {
  const float* x = (const float*)d_in[0];
  int B = in_sizes[0] / (3 * 224 * 224);       // 256 for the reference setup
  long nbn = (long)B * N_TOK;
  long nbk = (long)B * KEEP;

  // Workspace layout (all 4-byte slots)
  float* ws   = (float*)d_ws;
  float* tok  = ws;                                        // B*196*768
  float* sq   = tok + (long)B * N_TOK * CF;                // B*196
  float* dist = sq + nbn;                                  // B*196*196
  float* dens = dist + (long)B * N_TOK * N_TOK;            // B*196
  float* dmax = dens + nbn;                                // B
  float* scor = dmax + B;                                  // B*196
  int* index_down  = (int*)(scor + nbn);                   // B*49
  int* idx_cluster = index_down + nbk;                     // B*196

  long total_in = (long)B * 3 * 224 * 224;
  tokenize_kernel<<<(int)((total_in + 255) / 256), 256, 0, stream>>>(x, tok, total_in);
  sqnorm_kernel<<<(int)nbn, 256, 0, stream>>>(tok, sq);

  gemm_dist_kernel<<<B * 16, 256, 0, stream>>>(tok, sq, dist);

  density_kernel<<<(int)((nbn + 255) / 256), 256, 0, stream>>>(dist, dens, (int)nbn);
  distmax_kernel<<<B, 256, 0, stream>>>(dist, dmax);
  score_kernel<<<(int)((nbn + 255) / 256), 256, 0, stream>>>(dist, dens, dmax, scor, (int)nbn);
  topk_kernel<<<B, 32, 0, stream>>>(scor, index_down);
  assign_kernel<<<(int)((nbn + 255) / 256), 256, 0, stream>>>(dist, index_down, idx_cluster, (int)nbn);
  center_fix_kernel<<<(int)((nbk + 255) / 256), 256, 0, stream>>>(index_down, idx_cluster, (int)nbk);
  rep_kernel<<<(int)((nbk + 255) / 256), 256, 0, stream>>>(idx_cluster, (int*)d_out, (int)nbk);
}